// IDR_net_57904749085363
// MI455X (gfx1250) — compile-verified
//
#include <hip/hip_runtime.h>
#include <math.h>

// Problem constants (from reference)
#define Bsz   256
#define Tlen  1024
#define Hd    12
#define Kd    6
#define NROWS (Bsz * Tlen)   // 262144
#define G4    48             // 4*H gates
#define EPSV  1e-5f

typedef float v2f __attribute__((ext_vector_type(2)));
typedef float v8f __attribute__((ext_vector_type(8)));

__device__ __forceinline__ float sigm(float x) { return 1.0f / (1.0f + expf(-x)); }

// ---------------------------------------------------------------------------
// Kernel 1: LSTM input projection  G[row, 0:48] = X[row,0:12] @ Wih.T + (bih+bhh)
// Uses V_WMMA_F32_16X16X4_F32.  One wave per 16-row tile; N=48 -> 3 tiles of 16;
// K=12 -> 3 chunks of 4.  Fragment layouts per CDNA5 ISA 7.12.2 tables.
// ---------------------------------------------------------------------------
__global__ __launch_bounds__(256) void lstm_inproj_wmma(
    const float* __restrict__ X,     // [NROWS, 12]
    const float* __restrict__ Wih,   // [48, 12] (this layer)
    const float* __restrict__ bih,   // [48]
    const float* __restrict__ bhh,   // [48]
    float* __restrict__ G)           // [NROWS, 48]
{
    int wid   = threadIdx.x >> 5;
    int lane  = threadIdx.x & 31;
    int tile  = blockIdx.x * 8 + wid;
    int row0  = tile * 16;
    if (row0 >= NROWS) return;

    int m     = lane & 15;   // matrix row handled by this lane
    int khalf = lane >> 4;   // 0: K pair {0,1}; 1: K pair {2,3}

    // A fragments (16x4 f32): VGPR0 = K=khalf*2, VGPR1 = K=khalf*2+1
    v2f a[3];
#pragma unroll
    for (int kc = 0; kc < 3; ++kc) {
        int col = kc * 4 + khalf * 2;
        const float* xr = X + (size_t)(row0 + m) * Hd + col;
        a[kc].x = xr[0];
        a[kc].y = xr[1];
    }

#pragma unroll
    for (int nt = 0; nt < 3; ++nt) {
        int ncol = nt * 16 + (lane & 15);
        v8f c = {};  // f32 accumulator, 8 VGPRs
#pragma unroll
        for (int kc = 0; kc < 3; ++kc) {
            int k0 = kc * 4 + khalf * 2;
            // B is 4x16 (K x N) = Wih.T  ->  B[k][n] = Wih[n][k]
            v2f bf;
            bf.x = Wih[ncol * Hd + k0];
            bf.y = Wih[ncol * Hd + k0 + 1];
#if defined(__gfx1250__)
            // (neg_a, A, neg_b, B, c_mod, C, reuse_a, reuse_b)
            c = __builtin_amdgcn_wmma_f32_16x16x4_f32(
                    false, a[kc], false, bf, (short)0, c, false, false);
#else
            (void)bf;
#endif
        }
        float bias = bih[ncol] + bhh[ncol];
        // D layout: VGPR r holds row (r + khalf*8), column = lane&15
#pragma unroll
        for (int r = 0; r < 8; ++r) {
            int row = row0 + r + khalf * 8;
            G[(size_t)row * G4 + ncol] = c[r] + bias;
        }
    }
}

// ---------------------------------------------------------------------------
// Kernel 2: LSTM recurrence (the serial part).  4 threads per batch element:
// thread j computes gate block j (i/f/g/o), then updates 3 cell channels.
// h and gates exchanged through LDS (c kept in registers); Whh cached in LDS; next-step gates prefetched.
// ---------------------------------------------------------------------------
__global__ __launch_bounds__(128) void lstm_recur(
    const float* __restrict__ G,     // [NROWS, 48] input-projected gates
    const float* __restrict__ Whh,   // [48, 12]
    float* __restrict__ Xout)        // [NROWS, 12] layer output
{
    __shared__ float whh_s[G4 * Hd];
    __shared__ float hbuf[32 * Hd];
    __shared__ float gbuf[32 * G4];

    int tid  = threadIdx.x;        // 128 threads
    int elem = tid >> 2;           // 0..31 local batch element
    int j    = tid & 3;            // gate block: 0=i 1=f 2=g 3=o
    int b    = blockIdx.x * 32 + elem;

    for (int i = tid; i < G4 * Hd; i += 128) whh_s[i] = Whh[i];
    for (int i = tid; i < 32 * Hd; i += 128) hbuf[i] = 0.0f;
    float c0 = 0.0f, c1 = 0.0f, c2 = 0.0f;
    __syncthreads();

    const float* grow = G + (size_t)b * Tlen * G4 + 12 * j;
    float* xrow = Xout + (size_t)b * Tlen * Hd;

    for (int t = 0; t < Tlen; ++t) {
        const float* gp = grow + (size_t)t * G4;
        if (t + 1 < Tlen) __builtin_prefetch(gp + G4, 0, 3);  // global_prefetch_b8

        float hl[Hd];
#pragma unroll
        for (int d = 0; d < Hd; ++d) hl[d] = hbuf[elem * Hd + d];

#pragma unroll
        for (int i = 0; i < Hd; ++i) {
            int gi = 12 * j + i;
            float acc = gp[i];
            const float* wr = whh_s + gi * Hd;
#pragma unroll
            for (int d = 0; d < Hd; ++d) acc += wr[d] * hl[d];
            gbuf[elem * G4 + gi] = acc;
        }
        __syncthreads();

        int ch0 = 3 * j;
        float cc[3] = {c0, c1, c2};
#pragma unroll
        for (int q = 0; q < 3; ++q) {
            int ch = ch0 + q;
            float ir  = gbuf[elem * G4 + ch];
            float fr  = gbuf[elem * G4 + 12 + ch];
            float gr  = gbuf[elem * G4 + 24 + ch];
            float orr = gbuf[elem * G4 + 36 + ch];
            float cv  = sigm(fr) * cc[q] + sigm(ir) * tanhf(gr);
            cc[q] = cv;
            float hv = sigm(orr) * tanhf(cv);
            hbuf[elem * Hd + ch] = hv;
            xrow[(size_t)t * Hd + ch] = hv;
        }
        c0 = cc[0]; c1 = cc[1]; c2 = cc[2];
        __syncthreads();
    }
}

// ---------------------------------------------------------------------------
// Kernel 3: fused per-timestep MLP + BatchNorm(batch axis) + output mix.
// One block per timestep; thread = batch element, so BN stats are block-local.
// ---------------------------------------------------------------------------
__global__ __launch_bounds__(256) void mlp_fused(
    const float* __restrict__ X3, const float* __restrict__ MF,
    const float* __restrict__ fc1w, const float* __restrict__ fc1b,
    const float* __restrict__ g1, const float* __restrict__ b1,
    const float* __restrict__ fc2w, const float* __restrict__ fc2b,
    const float* __restrict__ g2, const float* __restrict__ b2,
    const float* __restrict__ fc3w, const float* __restrict__ fc3b,
    const float* __restrict__ WIp, float* __restrict__ out)
{
    __shared__ float s1w[64 * 12], s1b[64], sg1[64], sb1[64];
    __shared__ float s2w[128 * 64], s2b[128], sg2[128], sb2[128];
    __shared__ float s3w[6 * 128], s3b[6];
    __shared__ float swi[6 * 12 * 12];
    __shared__ float ps[8 * 64], psq[8 * 64];
    __shared__ float mu[64], rs[64];
    __shared__ float mu2[32], rs2[32];

    int tid = threadIdx.x;      // batch index 0..255
    int t   = blockIdx.x;       // timestep
    int lane = tid & 31, wid = tid >> 5;

    for (int i = tid; i < 64 * 12; i += 256)   s1w[i] = fc1w[i];
    for (int i = tid; i < 128 * 64; i += 256)  s2w[i] = fc2w[i];
    for (int i = tid; i < 6 * 128; i += 256)   s3w[i] = fc3w[i];
    for (int i = tid; i < 6 * 12 * 12; i += 256) swi[i] = WIp[i];
    if (tid < 64)  { s1b[tid] = fc1b[tid]; sg1[tid] = g1[tid]; sb1[tid] = b1[tid]; }
    if (tid < 128) { s2b[tid] = fc2b[tid]; sg2[tid] = g2[tid]; sb2[tid] = b2[tid]; }
    if (tid < 6)   s3b[tid] = fc3b[tid];
    __syncthreads();

    size_t row = (size_t)tid * Tlen + t;
    float x[12], mf[12];
#pragma unroll
    for (int d = 0; d < 12; ++d) { x[d] = X3[row * 12 + d]; mf[d] = MF[row * 12 + d]; }

    // fc1: 12 -> 64
    float h1[64];
#pragma unroll
    for (int o = 0; o < 64; ++o) {
        float acc = s1b[o];
#pragma unroll
        for (int d = 0; d < 12; ++d) acc += s1w[o * 12 + d] * x[d];
        h1[o] = acc;
    }
    // BN1 stats over batch (256 threads): wave shuffle + cross-wave LDS
    for (int c = 0; c < 64; ++c) {
        float v = h1[c], v2 = v * v;
        for (int off = 16; off; off >>= 1) {
            v  += __shfl_xor(v, off, 32);
            v2 += __shfl_xor(v2, off, 32);
        }
        if (lane == 0) { ps[wid * 64 + c] = v; psq[wid * 64 + c] = v2; }
    }
    __syncthreads();
    if (tid < 64) {
        float s = 0.0f, s2 = 0.0f;
        for (int w = 0; w < 8; ++w) { s += ps[w * 64 + tid]; s2 += psq[w * 64 + tid]; }
        float m  = s * (1.0f / 256.0f);
        float var = s2 * (1.0f / 256.0f) - m * m;
        mu[tid] = m; rs[tid] = rsqrtf(var + EPSV);
    }
    __syncthreads();
#pragma unroll
    for (int c = 0; c < 64; ++c) {
        float v = sg1[c] * (h1[c] - mu[c]) * rs[c] + sb1[c];
        h1[c] = v > 0.0f ? v : 0.0f;
    }
    __syncthreads();  // ps/psq reused below

    float lamb[6];
#pragma unroll
    for (int k = 0; k < 6; ++k) lamb[k] = s3b[k];

    // fc2 (64 -> 128) in 4 chunks of 32 + BN2 + ReLU + fc3 accumulation
    for (int ccu = 0; ccu < 4; ++ccu) {
        float h2[32];
        for (int o = 0; o < 32; ++o) {
            int O = ccu * 32 + o;
            float acc = s2b[O];
#pragma unroll
            for (int c = 0; c < 64; ++c) acc += s2w[O * 64 + c] * h1[c];
            h2[o] = acc;
        }
        for (int o = 0; o < 32; ++o) {
            float v = h2[o], v2 = v * v;
            for (int off = 16; off; off >>= 1) {
                v  += __shfl_xor(v, off, 32);
                v2 += __shfl_xor(v2, off, 32);
            }
            if (lane == 0) { ps[wid * 64 + o] = v; psq[wid * 64 + o] = v2; }
        }
        __syncthreads();
        if (tid < 32) {
            float s = 0.0f, s2 = 0.0f;
            for (int w = 0; w < 8; ++w) { s += ps[w * 64 + tid]; s2 += psq[w * 64 + tid]; }
            float m  = s * (1.0f / 256.0f);
            float var = s2 * (1.0f / 256.0f) - m * m;
            mu2[tid] = m; rs2[tid] = rsqrtf(var + EPSV);
        }
        __syncthreads();
        for (int o = 0; o < 32; ++o) {
            int O = ccu * 32 + o;
            float v = sg2[O] * (h2[o] - mu2[o]) * rs2[o] + sb2[O];
            v = v > 0.0f ? v : 0.0f;
#pragma unroll
            for (int k = 0; k < 6; ++k) lamb[k] += s3w[k * 128 + O] * v;
        }
        __syncthreads();
    }

    // out[b,t,:] = mfcc + sum_k lamb_k * (mfcc @ WI_k)
#pragma unroll
    for (int e = 0; e < 12; ++e) {
        float acc = mf[e];
#pragma unroll
        for (int k = 0; k < 6; ++k) {
            float pr = 0.0f;
#pragma unroll
            for (int d = 0; d < 12; ++d) pr += mf[d] * swi[(k * 12 + d) * 12 + e];
            acc += lamb[k] * pr;
        }
        out[row * 12 + e] = acc;
    }
}

// ---------------------------------------------------------------------------
extern "C" void kernel_launch(void* const* d_in, const int* in_sizes, int n_in,
                              void* d_out, int out_size, void* d_ws, size_t ws_size,
                              hipStream_t stream) {
    (void)in_sizes; (void)n_in; (void)out_size; (void)ws_size;
    const float* mfcc = (const float*)d_in[0];
    const float* Wih  = (const float*)d_in[1];   // [3,48,12]
    const float* Whh  = (const float*)d_in[2];   // [3,48,12]
    const float* bih  = (const float*)d_in[3];   // [3,48]
    const float* bhh  = (const float*)d_in[4];   // [3,48]
    const float* fc1w = (const float*)d_in[5];
    const float* fc1b = (const float*)d_in[6];
    const float* bn1g = (const float*)d_in[7];
    const float* bn1b = (const float*)d_in[8];
    const float* fc2w = (const float*)d_in[9];
    const float* fc2b = (const float*)d_in[10];
    const float* bn2g = (const float*)d_in[11];
    const float* bn2b = (const float*)d_in[12];
    const float* fc3w = (const float*)d_in[13];
    const float* fc3b = (const float*)d_in[14];
    const float* WIp  = (const float*)d_in[15];
    float* out = (float*)d_out;

    // Workspace: G gates [NROWS*48] + two ping-pong layer buffers [NROWS*12]
    float* ws = (float*)d_ws;
    float* G  = ws;
    float* xa = ws + (size_t)NROWS * G4;
    float* xb = xa + (size_t)NROWS * Hd;

    const float* xin = mfcc;
    float* dst[3] = {xa, xb, xa};
    for (int l = 0; l < 3; ++l) {
        lstm_inproj_wmma<<<NROWS / 16 / 8, 256, 0, stream>>>(
            xin, Wih + l * G4 * Hd, bih + l * G4, bhh + l * G4, G);
        lstm_recur<<<Bsz / 32, 128, 0, stream>>>(G, Whh + l * G4 * Hd, dst[l]);
        xin = dst[l];
    }
    mlp_fused<<<Tlen, 256, 0, stream>>>(xin, mfcc,
                                        fc1w, fc1b, bn1g, bn1b,
                                        fc2w, fc2b, bn2g, bn2b,
                                        fc3w, fc3b, WIp, out);
}